// ParallelCAttention_71073118814779
// MI455X (gfx1250) — compile-verified
//
#include <hip/hip_runtime.h>

// ---------------------------------------------------------------------------
// B=8, CIN=256, COUT=64, CQK=8, padded spatial 128x128 -> NSP=16384/batch.
// All GEMM-shaped work uses V_WMMA_F32_16X16X4_F32 (fp32, wave32, CDNA5).
// Shared operands staged into LDS with GLOBAL_LOAD_ASYNC_TO_LDS_* (ASYNCcnt).
// Problem is HBM-bound (~55 GFLOP vs ~0.6 GB traffic @ 23.3 TB/s), so fp32
// WMMA matches the fp32 reference with no throughput penalty.
// ---------------------------------------------------------------------------
#define BATCH 8
#define CIN   256
#define COUT  64
#define CQK   8
#define VDIM  128
#define NSP   16384          // 128*128
#define NEGV  (-1.0e30f)

typedef float v2f __attribute__((ext_vector_type(2)));
typedef float v8f __attribute__((ext_vector_type(8)));

__device__ __forceinline__ v8f wmma4(v2f a, v2f b, v8f c) {
  // D = A(16x4,f32) * B(4x16,f32) + C(16x16,f32)
  return __builtin_amdgcn_wmma_f32_16x16x4_f32(false, a, false, b, (short)0, c,
                                               false, false);
}

// Async global->LDS copies (CDNA5). LDS offset is relative to the workgroup's
// LDS base; our single __shared__ array starts at offset 0.
__device__ __forceinline__ void ldsAsyncB128(unsigned ldsOff, const void* g) {
  asm volatile("global_load_async_to_lds_b128 %0, %1, off"
               :: "v"(ldsOff), "v"((unsigned long long)(uintptr_t)g)
               : "memory");
}
__device__ __forceinline__ void ldsAsyncB32(unsigned ldsOff, const void* g) {
  asm volatile("global_load_async_to_lds_b32 %0, %1, off"
               :: "v"(ldsOff), "v"((unsigned long long)(uintptr_t)g)
               : "memory");
}
__device__ __forceinline__ void waitAsync0() {
#if __has_builtin(__builtin_amdgcn_s_wait_asynccnt)
  __builtin_amdgcn_s_wait_asynccnt(0);
#else
  asm volatile("s_wait_asynccnt 0x0" ::: "memory");
#endif
}

// Lane decomposition for one 16x16 tile per wave (wave32):
//   lane16 -> A row (M) / B,C column (N); half -> K group {0,1}|{2,3} (A/B),
//   row group M|M+8 (C/D).
#define TILE_LANES()                                        \
  const int lane   = threadIdx.x & 31;                      \
  const int wave   = threadIdx.x >> 5;                      \
  const int lane16 = lane & 15;                             \
  const int half   = lane >> 4;

// ---------------------------------------------------------------------------
// Kernel 1: y = dc_w (64x256) @ pad(x) + dc_b, also seeds d_out = y.
// Block = one 16-col N-tile, 4 waves = 4 M-tiles (share both panels via LDS).
// dc_w (64KB) + zero-padded B panel (256x16, 16KB) staged with async-to-LDS.
// Grid: 8192 blocks (8 b * 1024 n-tiles) x 128 thr.
// ---------------------------------------------------------------------------
__global__ void k_dcconv(const float* __restrict__ x,
                         const float* __restrict__ w,
                         const float* __restrict__ bias,
                         float* __restrict__ y,
                         float* __restrict__ out) {
  __shared__ float smem[64 * 256 + 256 * 16];   // sW | sB   (80 KB)
  float* sW = smem;                              // 64 x 256, LDS offset 0
  float* sB = smem + 64 * 256;                   // 256 x 16, LDS offset 65536
  const int tid = threadIdx.x;
  TILE_LANES();

  const int b  = blockIdx.x >> 10;
  const int n0 = (blockIdx.x & 1023) * 16;
  const int m0 = wave * 16;

  // Stage dc_w: 4096 x b128 chunks (32 per thread), async.
  for (int i = tid; i < 4096; i += 128)
    ldsAsyncB128((unsigned)(i * 16), w + (size_t)i * 4);

  // Zero the B panel (covers padding), then async-load interior elements.
  for (int i = tid; i < 4096; i += 128) sB[i] = 0.0f;
  __syncthreads();
  for (int i = tid; i < 4096; i += 128) {
    const int k = i >> 4, j = i & 15;
    const int n = n0 + j, vv = n >> 7, tt = n & 127;
    const bool interior =
        ((unsigned)(vv - 1) < 126u) && ((unsigned)(tt - 1) < 126u);
    if (interior)   // exec-masked async copy; masked-off lanes keep the zero
      ldsAsyncB32((unsigned)((64 * 256 + i) * 4),
                  x + (size_t)(b * CIN + k) * 15876 + (vv - 1) * 126 + (tt - 1));
  }
  waitAsync0();
  __syncthreads();

  // Branch-free WMMA loop, all operands from LDS.
  const int m = m0 + lane16;
  v8f acc = {};
  for (int k0 = 0; k0 < CIN; k0 += 4) {
    const int kk = k0 + half * 2;
    v2f a, bb;
    a.x  = sW[m * CIN + kk];
    a.y  = sW[m * CIN + kk + 1];
    bb.x = sB[kk * 16 + lane16];
    bb.y = sB[(kk + 1) * 16 + lane16];
    acc = wmma4(a, bb, acc);
  }
#pragma unroll
  for (int r = 0; r < 8; ++r) {
    const int row = m0 + r + half * 8;
    const float val = acc[r] + bias[row];
    const size_t idx = (size_t)(b * COUT + row) * NSP + n0 + lane16;
    y[idx]   = val;
    out[idx] = val;   // accumulator seed: out = y (+x1 +x2 added later)
  }
}

// ---------------------------------------------------------------------------
// Kernel 2: projection   outb(M x 16384) = w(M x 64) @ yb + bias, M in {8,64}.
// Weights async-staged to LDS, zero-padded so the M<mrows guard leaves the
// inner loop. Grid = (mtiles*1024/4) blocks x 128 thr.
// ---------------------------------------------------------------------------
__global__ void k_proj(const float* __restrict__ w,
                       const float* __restrict__ bias,
                       const float* __restrict__ yb,
                       float* __restrict__ outb,
                       int mrows) {
  __shared__ float sW[64 * 64];                  // 16 KB, LDS offset 0
  const int tid = threadIdx.x;
  TILE_LANES();

  for (int i = tid; i < 64 * 64; i += 128) sW[i] = 0.0f;
  __syncthreads();
  for (int i = tid; i < mrows * 16; i += 128)    // mrows*64/4 b128 chunks
    ldsAsyncB128((unsigned)(i * 16), w + (size_t)i * 4);
  waitAsync0();
  __syncthreads();

  const int tile = blockIdx.x * 4 + wave;
  const int m0 = (tile >> 10) * 16;
  const int n0 = (tile & 1023) * 16;
  const int m = m0 + lane16;
  const int n = n0 + lane16;

  v8f acc = {};
  for (int k0 = 0; k0 < COUT; k0 += 4) {
    const int kk = k0 + half * 2;
    v2f a, bb;
    a.x  = sW[m * COUT + kk];                    // zero-padded rows >= mrows
    a.y  = sW[m * COUT + kk + 1];
    bb.x = yb[(size_t)kk * NSP + n];
    bb.y = yb[(size_t)(kk + 1) * NSP + n];
    acc = wmma4(a, bb, acc);
  }
#pragma unroll
  for (int r = 0; r < 8; ++r) {
    const int row = m0 + r + half * 8;
    if (row < mrows)
      outb[(size_t)row * NSP + n0 + lane16] = acc[r] + bias[row];
  }
}

// ---------------------------------------------------------------------------
// Kernel 3: axial energies (K=8). mode=0: E_H per fixed t (diag masked);
// mode=1: E_W per fixed v.  E row layout: [(v*128+t)*256 + {u | 128+s}]
// Grid: dim3(16, 128) x 128 thr.
// ---------------------------------------------------------------------------
__global__ void k_energy(const float* __restrict__ qb,
                         const float* __restrict__ kb,
                         float* __restrict__ Eb, int mode) {
  TILE_LANES();
  const int fixed = blockIdx.y;                 // t (mode 0) or v (mode 1)
  const int tile = blockIdx.x * 4 + wave;       // 64 tiles (8x8)
  const int m0 = (tile >> 3) * 16;
  const int n0 = (tile & 7) * 16;
  const int m = m0 + lane16;
  const int n = n0 + lane16;

  v8f acc = {};
  for (int k0 = 0; k0 < CQK; k0 += 4) {
    const int kk = k0 + half * 2;
    v2f a, bb;
    if (mode == 1) {  // W: A[t,c]=q[c,v,t]  B[c,s]=k[c,v,s]
      a.x  = qb[(size_t)kk * NSP + fixed * VDIM + m];
      a.y  = qb[(size_t)(kk + 1) * NSP + fixed * VDIM + m];
      bb.x = kb[(size_t)kk * NSP + fixed * VDIM + n];
      bb.y = kb[(size_t)(kk + 1) * NSP + fixed * VDIM + n];
    } else {          // H: A[v,c]=q[c,v,t]  B[c,u]=k[c,u,t]
      a.x  = qb[(size_t)kk * NSP + m * VDIM + fixed];
      a.y  = qb[(size_t)(kk + 1) * NSP + m * VDIM + fixed];
      bb.x = kb[(size_t)kk * NSP + n * VDIM + fixed];
      bb.y = kb[(size_t)(kk + 1) * NSP + n * VDIM + fixed];
    }
    acc = wmma4(a, bb, acc);
  }
#pragma unroll
  for (int r = 0; r < 8; ++r) {
    const int row = m0 + r + half * 8;
    const int col = n0 + lane16;
    if (mode == 1) {
      Eb[(size_t)(fixed * VDIM + row) * 256 + 128 + col] = acc[r];
    } else {
      float val = (row == col) ? NEGV : acc[r];
      Eb[(size_t)(row * VDIM + fixed) * 256 + col] = val;
    }
  }
}

// ---------------------------------------------------------------------------
// Kernel 4: softmax over 256 per row; 16384 rows; 1 wave/row.
// ---------------------------------------------------------------------------
__global__ void k_softmax256(float* __restrict__ Eb) {
  const int lane = threadIdx.x & 31;
  const int row = blockIdx.x * 8 + (threadIdx.x >> 5);
  float* p = Eb + (size_t)row * 256;
  float v[8];
  float mx = -3.402823e38f;
#pragma unroll
  for (int j = 0; j < 8; ++j) { v[j] = p[lane + j * 32]; mx = fmaxf(mx, v[j]); }
#pragma unroll
  for (int off = 16; off > 0; off >>= 1) mx = fmaxf(mx, __shfl_xor(mx, off, 32));
  float sum = 0.0f;
#pragma unroll
  for (int j = 0; j < 8; ++j) { v[j] = __expf(v[j] - mx); sum += v[j]; }
#pragma unroll
  for (int off = 16; off > 0; off >>= 1) sum += __shfl_xor(sum, off, 32);
  const float inv = 1.0f / sum;
#pragma unroll
  for (int j = 0; j < 8; ++j) p[lane + j * 32] = v[j] * inv;
}

// ---------------------------------------------------------------------------
// Kernel 5: out_H apply per fixed t:  out[c,vv] += g * sum_u v[c,u,t]*A_H[vv,u]
// Grid: dim3(8, 128) x 128 thr.  K = 128.
// ---------------------------------------------------------------------------
__global__ void k_outH(const float* __restrict__ vb,
                       const float* __restrict__ Eb,
                       float* __restrict__ outb,
                       const float* __restrict__ gamma) {
  TILE_LANES();
  const int t = blockIdx.y;
  const int tile = blockIdx.x * 4 + wave;       // 32 tiles (4 M x 8 N)
  const int m0 = (tile >> 3) * 16;
  const int n0 = (tile & 7) * 16;
  const int m = m0 + lane16;
  const int n = n0 + lane16;                    // target row vv

  v8f acc = {};
  for (int k0 = 0; k0 < VDIM; k0 += 4) {
    const int kk = k0 + half * 2;
    v2f a, bb;
    a.x  = vb[(size_t)m * NSP + kk * VDIM + t];
    a.y  = vb[(size_t)m * NSP + (kk + 1) * VDIM + t];
    bb.x = Eb[(size_t)(n * VDIM + t) * 256 + kk];
    bb.y = Eb[(size_t)(n * VDIM + t) * 256 + kk + 1];
    acc = wmma4(a, bb, acc);
  }
  const float g = gamma[0];
#pragma unroll
  for (int r = 0; r < 8; ++r) {
    const int row = m0 + r + half * 8;
    const int col = n0 + lane16;
    outb[(size_t)row * NSP + col * VDIM + t] += g * acc[r];
  }
}

// ---------------------------------------------------------------------------
// Kernel 6: out_W apply per fixed v:  out[c,t] += g * sum_s v[c,v,s]*A_W[t,s]
// ---------------------------------------------------------------------------
__global__ void k_outW(const float* __restrict__ vb,
                       const float* __restrict__ Eb,
                       float* __restrict__ outb,
                       const float* __restrict__ gamma) {
  TILE_LANES();
  const int v = blockIdx.y;
  const int tile = blockIdx.x * 4 + wave;
  const int m0 = (tile >> 3) * 16;
  const int n0 = (tile & 7) * 16;
  const int m = m0 + lane16;
  const int n = n0 + lane16;                    // target t

  v8f acc = {};
  for (int k0 = 0; k0 < VDIM; k0 += 4) {
    const int kk = k0 + half * 2;
    v2f a, bb;
    a.x  = vb[(size_t)m * NSP + v * VDIM + kk];
    a.y  = vb[(size_t)m * NSP + v * VDIM + kk + 1];
    bb.x = Eb[(size_t)(v * VDIM + n) * 256 + 128 + kk];
    bb.y = Eb[(size_t)(v * VDIM + n) * 256 + 128 + kk + 1];
    acc = wmma4(a, bb, acc);
  }
  const float g = gamma[0];
#pragma unroll
  for (int r = 0; r < 8; ++r) {
    const int row = m0 + r + half * 8;
    const int col = n0 + lane16;
    outb[(size_t)row * NSP + v * VDIM + col] += g * acc[r];
  }
}

// ---------------------------------------------------------------------------
// Kernel 7: channel-attn energies per fixed v:  S[c,d] = sum_t qc[c,t]kc[d,t]
// Grid: dim3(4, 128) x 128 thr.  K = 128.
// ---------------------------------------------------------------------------
__global__ void k_cattn(const float* __restrict__ qcb,
                        const float* __restrict__ kcb,
                        float* __restrict__ Sb) {
  TILE_LANES();
  const int v = blockIdx.y;
  const int tile = blockIdx.x * 4 + wave;       // 16 tiles (4x4)
  const int m0 = (tile >> 2) * 16;
  const int n0 = (tile & 3) * 16;
  const int m = m0 + lane16;
  const int n = n0 + lane16;

  v8f acc = {};
  for (int k0 = 0; k0 < VDIM; k0 += 4) {
    const int kk = k0 + half * 2;
    v2f a, bb;
    a.x  = qcb[(size_t)m * NSP + v * VDIM + kk];
    a.y  = qcb[(size_t)m * NSP + v * VDIM + kk + 1];
    bb.x = kcb[(size_t)n * NSP + v * VDIM + kk];
    bb.y = kcb[(size_t)n * NSP + v * VDIM + kk + 1];
    acc = wmma4(a, bb, acc);
  }
#pragma unroll
  for (int r = 0; r < 8; ++r) {
    const int row = m0 + r + half * 8;
    Sb[(size_t)v * 4096 + row * 64 + n0 + lane16] = acc[r];
  }
}

// ---------------------------------------------------------------------------
// Kernel 8: softmax over 64 per row; 8192 rows (v*64+c); 1 wave/row.
// ---------------------------------------------------------------------------
__global__ void k_softmax64(float* __restrict__ Sb) {
  const int lane = threadIdx.x & 31;
  const int row = blockIdx.x * 8 + (threadIdx.x >> 5);
  float* p = Sb + (size_t)row * 64;
  float v0 = p[lane], v1 = p[lane + 32];
  float mx = fmaxf(v0, v1);
#pragma unroll
  for (int off = 16; off > 0; off >>= 1) mx = fmaxf(mx, __shfl_xor(mx, off, 32));
  v0 = __expf(v0 - mx); v1 = __expf(v1 - mx);
  float sum = v0 + v1;
#pragma unroll
  for (int off = 16; off > 0; off >>= 1) sum += __shfl_xor(sum, off, 32);
  const float inv = 1.0f / sum;
  p[lane] = v0 * inv; p[lane + 32] = v1 * inv;
}

// ---------------------------------------------------------------------------
// Kernel 9: channel-attn apply per fixed v: av[c,t] = sum_d S[v,c,d]*vc[d,v,t]
// Grid: dim3(8, 128) x 128 thr.  K = 64.
// ---------------------------------------------------------------------------
__global__ void k_cav(const float* __restrict__ Sb,
                      const float* __restrict__ vcb,
                      float* __restrict__ avb) {
  TILE_LANES();
  const int v = blockIdx.y;
  const int tile = blockIdx.x * 4 + wave;       // 32 tiles (4 M x 8 N)
  const int m0 = (tile >> 3) * 16;
  const int n0 = (tile & 7) * 16;
  const int m = m0 + lane16;
  const int n = n0 + lane16;

  v8f acc = {};
  for (int k0 = 0; k0 < COUT; k0 += 4) {
    const int kk = k0 + half * 2;
    v2f a, bb;
    a.x  = Sb[(size_t)v * 4096 + m * 64 + kk];
    a.y  = Sb[(size_t)v * 4096 + m * 64 + kk + 1];
    bb.x = vcb[(size_t)kk * NSP + v * VDIM + n];
    bb.y = vcb[(size_t)(kk + 1) * NSP + v * VDIM + n];
    acc = wmma4(a, bb, acc);
  }
#pragma unroll
  for (int r = 0; r < 8; ++r) {
    const int row = m0 + r + half * 8;
    avb[(size_t)row * NSP + v * VDIM + n0 + lane16] = acc[r];
  }
}

// ---------------------------------------------------------------------------
// Kernel 10: 3x3 conv as 9 shifted K=64 WMMA GEMMs; out += sigma * conv.
// Block = one M-tile x 4 N-tiles; 16x64x9 weight slab (36 KB) async-staged
// to LDS. Halo handled by mask-multiply (branch-free inner loop).
// Grid: 1024 blocks (4 mt * 256 nblk) x 128 thr.
// ---------------------------------------------------------------------------
__global__ void k_conv3x3(const float* __restrict__ aw,
                          const float* __restrict__ avb,
                          float* __restrict__ outb,
                          const float* __restrict__ sigma) {
  __shared__ float sAW[16 * 64 * 9];             // 9216 floats, 36 KB
  const int tid = threadIdx.x;
  TILE_LANES();

  const int m0 = (blockIdx.x >> 8) * 16;
  const int n0 = ((blockIdx.x & 255) * 4 + wave) * 16;

  // Stage rows m0..m0+15 of av_w (contiguous 16*576 floats) via async b128.
  for (int i = tid; i < 2304; i += 128)          // 9216/4 chunks
    ldsAsyncB128((unsigned)(i * 16), aw + (size_t)m0 * 576 + (size_t)i * 4);
  waitAsync0();
  __syncthreads();

  const int n = n0 + lane16;
  const int vv = n >> 7, tt = n & 127;

  v8f acc = {};
#pragma unroll
  for (int ij = 0; ij < 9; ++ij) {
    const int di = ij / 3 - 1, dj = ij % 3 - 1;
    const int v2 = vv + di, t2 = tt + dj;
    const bool inb = ((unsigned)v2 < 128u) && ((unsigned)t2 < 128u);
    const float msk = inb ? 1.0f : 0.0f;
    const int boff = inb ? (v2 * VDIM + t2) : 0; // clamped-safe offset
    for (int k0 = 0; k0 < COUT; k0 += 4) {
      const int kk = k0 + half * 2;
      v2f a, bb;
      a.x  = sAW[(lane16 * COUT + kk) * 9 + ij];
      a.y  = sAW[(lane16 * COUT + kk + 1) * 9 + ij];
      bb.x = msk * avb[(size_t)kk * NSP + boff];
      bb.y = msk * avb[(size_t)(kk + 1) * NSP + boff];
      acc = wmma4(a, bb, acc);
    }
  }
  const float s = sigma[0];
#pragma unroll
  for (int r = 0; r < 8; ++r) {
    const int row = m0 + r + half * 8;
    outb[(size_t)row * NSP + n0 + lane16] += s * acc[r];
  }
}

// ---------------------------------------------------------------------------
// Host launcher. Workspace (floats): y 8388608 | qb,kb 131072 ea |
// vb,qcb,kcb,vcb 1048576 ea | Eb 4194304 | Sb 524288 | avb 1048576
// Total = 18,612,224 floats ~ 71 MB.
// ---------------------------------------------------------------------------
extern "C" void kernel_launch(void* const* d_in, const int* in_sizes, int n_in,
                              void* d_out, int out_size, void* d_ws, size_t ws_size,
                              hipStream_t stream) {
  const float* x     = (const float*)d_in[0];
  const float* dc_w  = (const float*)d_in[1];
  const float* dc_b  = (const float*)d_in[2];
  const float* q_w   = (const float*)d_in[3];
  const float* q_b   = (const float*)d_in[4];
  const float* k_w   = (const float*)d_in[5];
  const float* k_b   = (const float*)d_in[6];
  const float* v_w   = (const float*)d_in[7];
  const float* v_b   = (const float*)d_in[8];
  const float* gamma = (const float*)d_in[9];
  const float* cq_w  = (const float*)d_in[10];
  const float* cq_b  = (const float*)d_in[11];
  const float* ck_w  = (const float*)d_in[12];
  const float* ck_b  = (const float*)d_in[13];
  const float* cv_w  = (const float*)d_in[14];
  const float* cv_b  = (const float*)d_in[15];
  const float* av_w  = (const float*)d_in[16];
  const float* sigma = (const float*)d_in[17];

  float* ws = (float*)d_ws;
  float* y   = ws;                       // 8388608
  float* qb  = y   + 8388608;            // 131072
  float* kb  = qb  + 131072;             // 131072
  float* vb  = kb  + 131072;             // 1048576
  float* qcb = vb  + 1048576;            // 1048576
  float* kcb = qcb + 1048576;            // 1048576
  float* vcb = kcb + 1048576;            // 1048576
  float* Eb  = vcb + 1048576;            // 4194304
  float* Sb  = Eb  + 4194304;            // 524288
  float* avb = Sb  + 524288;             // 1048576

  float* out = (float*)d_out;

  // Stage 1: y = dc conv (async-LDS staged WMMA GEMM), seeds out = y.
  k_dcconv<<<8192, 128, 0, stream>>>(x, dc_w, dc_b, y, out);

  for (int b = 0; b < BATCH; ++b) {
    const float* yb = y + (size_t)b * COUT * NSP;
    float* ob = out + (size_t)b * COUT * NSP;

    // Stage 2: projections from y (per batch, WMMA GEMMs).
    k_proj<<<256,  128, 0, stream>>>(q_w,  q_b,  yb, qb,  CQK);
    k_proj<<<256,  128, 0, stream>>>(k_w,  k_b,  yb, kb,  CQK);
    k_proj<<<1024, 128, 0, stream>>>(v_w,  v_b,  yb, vb,  COUT);
    k_proj<<<1024, 128, 0, stream>>>(cq_w, cq_b, yb, qcb, COUT);
    k_proj<<<1024, 128, 0, stream>>>(ck_w, ck_b, yb, kcb, COUT);
    k_proj<<<1024, 128, 0, stream>>>(cv_w, cv_b, yb, vcb, COUT);

    // Stage 3: axial attention (H+W energies -> joint softmax -> apply).
    k_energy<<<dim3(16, 128), 128, 0, stream>>>(qb, kb, Eb, 0);   // E_H
    k_energy<<<dim3(16, 128), 128, 0, stream>>>(qb, kb, Eb, 1);   // E_W
    k_softmax256<<<2048, 256, 0, stream>>>(Eb);
    k_outH<<<dim3(8, 128), 128, 0, stream>>>(vb, Eb, ob, gamma);
    k_outW<<<dim3(8, 128), 128, 0, stream>>>(vb, Eb, ob, gamma);

    // Stage 4: channel attention.
    k_cattn<<<dim3(4, 128), 128, 0, stream>>>(qcb, kcb, Sb);
    k_softmax64<<<1024, 256, 0, stream>>>(Sb);
    k_cav<<<dim3(8, 128), 128, 0, stream>>>(Sb, vcb, avb);

    // Stage 5: 3x3 conv of av (9 shifted WMMA GEMMs), out += sigma * conv.
    k_conv3x3<<<1024, 128, 0, stream>>>(av_w, avb, ob, sigma);
  }

  (void)in_sizes; (void)n_in; (void)out_size; (void)ws_size;
}